// ExternalAttention_90383291777576
// MI455X (gfx1250) — compile-verified
//
#include <hip/hip_runtime.h>

// ---------------------------------------------------------------------------
// ExternalAttention on MI455X (gfx1250, wave32, WMMA).
// Memory-bound (17.2 GFLOP vs ~290MB min HBM traffic -> ~12.5us @ 23.3TB/s),
// so keep fp32 end-to-end and use V_WMMA_F32_16X16X4_F32 for both GEMMs.
// BN is folded into GEMM1's weights; softmax/head-norm fused in place on q.
// GEMM waves use 2x2 fragment tiles (32Mx32N) -> block tile 64Mx128N, which
// halves B-matrix L2 traffic vs a 32M block and feeds 4 WMMAs per 4 loads.
// ---------------------------------------------------------------------------

typedef float v2f __attribute__((ext_vector_type(2)));
typedef float v8f __attribute__((ext_vector_type(8)));

constexpr int kB    = 8;
constexpr int kCIN  = 512;
constexpr int kCI   = 256;   // C_inter
constexpr int kCOUT = 512;
constexpr int kHW   = 4096;  // 64*64
constexpr int kNH   = 8;
constexpr int kDH   = kCI / kNH;  // 32

// ---------------------------------------------------------------------------
// K1: per-channel BN stats -> affine fold coefficients
//     a[c] = gamma*rsqrt(var+eps), b[c] = beta - mean*a[c]
// ---------------------------------------------------------------------------
__global__ __launch_bounds__(256) void bn_stats_kernel(
    const float* __restrict__ x, const float* __restrict__ gamma,
    const float* __restrict__ beta, float* __restrict__ ca,
    float* __restrict__ cb) {
  __shared__ float s1[256];
  __shared__ float s2[256];
  const int c = blockIdx.x;
  const int t = threadIdx.x;
  float sum = 0.f, ssq = 0.f;
  for (int e = t; e < kB * kHW; e += 256) {
    const int b = e >> 12;
    const int s = e & (kHW - 1);
    const float v = x[(size_t)b * kCIN * kHW + (size_t)c * kHW + s];
    sum += v;
    ssq += v * v;
  }
  s1[t] = sum;
  s2[t] = ssq;
  __syncthreads();
  for (int off = 128; off > 0; off >>= 1) {
    if (t < off) {
      s1[t] += s1[t + off];
      s2[t] += s2[t + off];
    }
    __syncthreads();
  }
  if (t == 0) {
    const float inv_n = 1.f / (float)(kB * kHW);
    const float mean = s1[0] * inv_n;
    const float var = s2[0] * inv_n - mean * mean;
    const float a = gamma[c] * rsqrtf(var + 1e-5f);
    ca[c] = a;
    cb[c] = beta[c] - mean * a;
  }
}

// ---------------------------------------------------------------------------
// K2: fold BN into weights: kp[i,c] = k[i,c]*a[c]; bias[i] = sum_c k[i,c]*b[c]
// ---------------------------------------------------------------------------
__global__ __launch_bounds__(256) void fold_k_kernel(
    const float* __restrict__ k, const float* __restrict__ ca,
    const float* __restrict__ cb, float* __restrict__ kp,
    float* __restrict__ bias) {
  __shared__ float s1[256];
  const int i = blockIdx.x;
  const int t = threadIdx.x;
  float acc = 0.f;
  for (int c = t; c < kCIN; c += 256) {
    const float kv = k[i * kCIN + c];
    kp[i * kCIN + c] = kv * ca[c];
    acc += kv * cb[c];
  }
  s1[t] = acc;
  __syncthreads();
  for (int off = 128; off > 0; off >>= 1) {
    if (t < off) s1[t] += s1[t + off];
    __syncthreads();
  }
  if (t == 0) bias[i] = s1[0];
}

// ---------------------------------------------------------------------------
// WMMA helper: D = A(16x4) * B(4x16) + C, full fp32 via V_WMMA_F32_16X16X4_F32
// A frag (lane L): L<16 -> {A[L][k0], A[L][k0+1]},  L>=16 -> {A[L-16][k0+2],..}
// B frag (lane L): L<16 -> {B[k0][n0+L], B[k0+1][n0+L]}, L>=16 -> K+2 half
// ---------------------------------------------------------------------------
__device__ __forceinline__ v8f wmma_f32_k4(v2f a, v2f b, v8f c) {
  return __builtin_amdgcn_wmma_f32_16x16x4_f32(false, a, false, b, (short)0, c,
                                               false, false);
}

// ---------------------------------------------------------------------------
// K3: GEMM1  q[b,i,s] = sum_c kp[i,c]*x[b,c,s] + bias[i]
//     8 waves arranged 2(M)x4(N); wave tile 32Mx32N (2x2 frags, 4 acc);
//     block tile 64Mx128N.  grid = (4096/128, 256/64, 8)
// ---------------------------------------------------------------------------
__global__ __launch_bounds__(256) void gemm1_kernel(
    const float* __restrict__ x, const float* __restrict__ kp,
    const float* __restrict__ bias, float* __restrict__ q) {
  const int tid = threadIdx.x;
  const int wave = tid >> 5;
  const int lane = tid & 31;
  const int wm = wave & 1;
  const int wn = wave >> 1;
  const int nl = lane & 15;    // M index for A-frag, N index for B/C-frag
  const int half = lane >> 4;  // K-pair select (A/B), M+8 select (C/D)
  const int m0 = blockIdx.y * 64 + wm * 32;
  const int n0 = blockIdx.x * 128 + wn * 32;
  const size_t bz = blockIdx.z;

  const float* __restrict__ xb = x + bz * (size_t)kCIN * kHW;
  const float* __restrict__ arow0 = kp + (size_t)(m0 + nl) * kCIN;
  const float* __restrict__ arow1 = arow0 + (size_t)16 * kCIN;

  v8f acc00 = {}, acc01 = {}, acc10 = {}, acc11 = {};
  for (int c0 = 0; c0 < kCIN; c0 += 4) {
    const int c = c0 + half * 2;
    v2f a0, a1;
    a0.x = arow0[c];
    a0.y = arow0[c + 1];
    a1.x = arow1[c];
    a1.y = arow1[c + 1];
    const float* __restrict__ xp = xb + (size_t)c * kHW + n0 + nl;
    if ((c0 & 31) == 0)  // stream x tiles ahead into L2 (global_prefetch_b8)
      __builtin_prefetch(xp + (size_t)32 * kHW, 0, 1);
    v2f b0, b1;
    b0.x = xp[0];
    b0.y = xp[kHW];
    b1.x = xp[16];
    b1.y = xp[kHW + 16];
    acc00 = wmma_f32_k4(a0, b0, acc00);
    acc01 = wmma_f32_k4(a0, b1, acc01);
    acc10 = wmma_f32_k4(a1, b0, acc10);
    acc11 = wmma_f32_k4(a1, b1, acc11);
  }
  float* __restrict__ qb = q + bz * (size_t)kCI * kHW;
  for (int r = 0; r < 8; ++r) {
    const int ma = m0 + half * 8 + r;
    const int mb = ma + 16;
    const float bia = bias[ma];
    const float bib = bias[mb];
    qb[(size_t)ma * kHW + n0 + nl] = acc00[r] + bia;
    qb[(size_t)ma * kHW + n0 + 16 + nl] = acc01[r] + bia;
    qb[(size_t)mb * kHW + n0 + nl] = acc10[r] + bib;
    qb[(size_t)mb * kHW + n0 + 16 + nl] = acc11[r] + bib;
  }
}

// ---------------------------------------------------------------------------
// K4: per-(b,i) softmax stats over the 4096 spatial positions.
//     Row lives in registers (16 vals/thread) -> single pass over q.
// ---------------------------------------------------------------------------
__global__ __launch_bounds__(256) void softmax_stats_kernel(
    const float* __restrict__ q, float* __restrict__ rowmax,
    float* __restrict__ rowinvz) {
  __shared__ float sred[256];
  const int row = blockIdx.x;  // b*kCI + i
  const int t = threadIdx.x;
  const float* __restrict__ qr = q + (size_t)row * kHW;
  float vals[16];
  float mx = -3.4e38f;
  for (int j = 0; j < 16; ++j) {
    vals[j] = qr[t + j * 256];
    mx = fmaxf(mx, vals[j]);
  }
  sred[t] = mx;
  __syncthreads();
  for (int off = 128; off > 0; off >>= 1) {
    if (t < off) sred[t] = fmaxf(sred[t], sred[t + off]);
    __syncthreads();
  }
  mx = sred[0];
  __syncthreads();
  float se = 0.f;
  for (int j = 0; j < 16; ++j) se += __expf(vals[j] - mx);
  sred[t] = se;
  __syncthreads();
  for (int off = 128; off > 0; off >>= 1) {
    if (t < off) sred[t] += sred[t + off];
    __syncthreads();
  }
  if (t == 0) {
    rowmax[row] = mx;
    rowinvz[row] = 1.f / sred[0];
  }
}

// ---------------------------------------------------------------------------
// K5: qf[b,i,s] = p / (sum_{i' in head} p' + 1e-6), p = exp(q-max)*invZ.
//     One thread per s; the 32 head-channels held in registers; in-place on q.
//     grid = kB * kNH * (kHW/256)
// ---------------------------------------------------------------------------
__global__ __launch_bounds__(256) void act_dn_kernel(
    float* __restrict__ q, const float* __restrict__ rowmax,
    const float* __restrict__ rowinvz) {
  const int chunk = blockIdx.x & 15;
  const int h = (blockIdx.x >> 4) & (kNH - 1);
  const int b = blockIdx.x >> 7;
  const int s = chunk * 256 + threadIdx.x;
  const int base_i = h * kDH;
  const int rbase = b * kCI + base_i;
  const size_t qoff = (size_t)b * kCI * kHW + (size_t)base_i * kHW + s;

  float p[kDH];
  float denom = 0.f;
  for (int j = 0; j < kDH; ++j) {
    const float val =
        __expf(q[qoff + (size_t)j * kHW] - rowmax[rbase + j]) * rowinvz[rbase + j];
    p[j] = val;
    denom += val;
  }
  const float rd = 1.f / (denom + 1e-6f);
  for (int j = 0; j < kDH; ++j) q[qoff + (size_t)j * kHW] = p[j] * rd;
}

// ---------------------------------------------------------------------------
// K6: GEMM2  out[b,o,s] = sum_i v[o,i]*qf[b,i,s].  Same tiling as GEMM1.
//     grid = (4096/128, 512/64, 8)
// ---------------------------------------------------------------------------
__global__ __launch_bounds__(256) void gemm2_kernel(
    const float* __restrict__ qf, const float* __restrict__ v,
    float* __restrict__ out) {
  const int tid = threadIdx.x;
  const int wave = tid >> 5;
  const int lane = tid & 31;
  const int wm = wave & 1;
  const int wn = wave >> 1;
  const int nl = lane & 15;
  const int half = lane >> 4;
  const int m0 = blockIdx.y * 64 + wm * 32;
  const int n0 = blockIdx.x * 128 + wn * 32;
  const size_t bz = blockIdx.z;

  const float* __restrict__ qb = qf + bz * (size_t)kCI * kHW;
  const float* __restrict__ arow0 = v + (size_t)(m0 + nl) * kCI;
  const float* __restrict__ arow1 = arow0 + (size_t)16 * kCI;

  v8f acc00 = {}, acc01 = {}, acc10 = {}, acc11 = {};
  for (int c0 = 0; c0 < kCI; c0 += 4) {
    const int c = c0 + half * 2;
    v2f a0, a1;
    a0.x = arow0[c];
    a0.y = arow0[c + 1];
    a1.x = arow1[c];
    a1.y = arow1[c + 1];
    const float* __restrict__ qp = qb + (size_t)c * kHW + n0 + nl;
    if ((c0 & 31) == 0)
      __builtin_prefetch(qp + (size_t)32 * kHW, 0, 1);
    v2f b0, b1;
    b0.x = qp[0];
    b0.y = qp[kHW];
    b1.x = qp[16];
    b1.y = qp[kHW + 16];
    acc00 = wmma_f32_k4(a0, b0, acc00);
    acc01 = wmma_f32_k4(a0, b1, acc01);
    acc10 = wmma_f32_k4(a1, b0, acc10);
    acc11 = wmma_f32_k4(a1, b1, acc11);
  }
  float* __restrict__ ob = out + bz * (size_t)kCOUT * kHW;
  for (int r = 0; r < 8; ++r) {
    const int ma = m0 + half * 8 + r;
    const int mb = ma + 16;
    ob[(size_t)ma * kHW + n0 + nl] = acc00[r];
    ob[(size_t)ma * kHW + n0 + 16 + nl] = acc01[r];
    ob[(size_t)mb * kHW + n0 + nl] = acc10[r];
    ob[(size_t)mb * kHW + n0 + 16 + nl] = acc11[r];
  }
}

// ---------------------------------------------------------------------------
// Launch
// ---------------------------------------------------------------------------
extern "C" void kernel_launch(void* const* d_in, const int* in_sizes, int n_in,
                              void* d_out, int out_size, void* d_ws,
                              size_t ws_size, hipStream_t stream) {
  const float* x = (const float*)d_in[0];      // [8,512,64,64]
  const float* k = (const float*)d_in[1];      // [256,512]
  const float* v = (const float*)d_in[2];      // [512,256]
  const float* gamma = (const float*)d_in[3];  // [512]
  const float* beta = (const float*)d_in[4];   // [512]
  float* out = (float*)d_out;                  // [8,512,64,64]

  // Workspace carve-up (floats): q (in-place reused for qf), folded k,
  // BN coeffs, bias, softmax row stats.
  float* ws = (float*)d_ws;
  float* q = ws;                           // 8*256*4096
  float* kp = q + (size_t)kB * kCI * kHW;  // 256*512
  float* ca = kp + (size_t)kCI * kCIN;     // 512
  float* cb = ca + kCIN;                   // 512
  float* bias = cb + kCIN;                 // 256
  float* rowmax = bias + kCI;              // 8*256
  float* rowinvz = rowmax + kB * kCI;      // 8*256

  bn_stats_kernel<<<kCIN, 256, 0, stream>>>(x, gamma, beta, ca, cb);
  fold_k_kernel<<<kCI, 256, 0, stream>>>(k, ca, cb, kp, bias);
  gemm1_kernel<<<dim3(kHW / 128, kCI / 64, kB), 256, 0, stream>>>(x, kp, bias,
                                                                  q);
  softmax_stats_kernel<<<kB * kCI, 256, 0, stream>>>(q, rowmax, rowinvz);
  act_dn_kernel<<<kB * kNH * (kHW / 256), 256, 0, stream>>>(q, rowmax, rowinvz);
  gemm2_kernel<<<dim3(kHW / 128, kCOUT / 64, kB), 256, 0, stream>>>(q, v, out);
}